// QDense_14302241096057
// MI455X (gfx1250) — compile-verified
//
#include <hip/hip_runtime.h>

// Problem constants (from reference: D=32, M=64, B=16)
#define DD 32
#define MM 64
#define NN 1024          // D*D
#define FF 2080          // M*(M+1)/2  (divisible by 16 -> 130 tiles)
#define FT 130           // FF / 16
#define BB 16
#define TS 1028          // padded LDS row stride in floats (1028 % 64 == 4 -> 16 rows on distinct banks)

typedef float v2f __attribute__((ext_vector_type(2)));
typedef float v8f __attribute__((ext_vector_type(8)));

// D = A(16x4 f32) * B(4x16 f32) + C(16x16 f32), full fp32 precision.
__device__ __forceinline__ v8f wmma4(v2f a, v2f b, v8f c) {
  // 8 args: (neg_a, A, neg_b, B, c_mod, C, reuse_a, reuse_b)
  return __builtin_amdgcn_wmma_f32_16x16x4_f32(false, a, false, b, (short)0, c,
                                               false, false);
}

__device__ __forceinline__ v2f ld2(const float* p) { return *(const v2f*)p; }

// ---------------------------------------------------------------------------
// Kernel 1: build u_evolve = amplitudes^T, stored row-major A[f][i], i = x*D+y.
//   amp(f; x,y) = Up[j,x]*Uq[k,y] + Up[k,x]*Uq[j,y], scaled 1/sqrt(2) if j==k
// ---------------------------------------------------------------------------
__global__ void slos_amp_kernel(const float* __restrict__ U_re,
                                const float* __restrict__ U_im,
                                float* __restrict__ A_re,
                                float* __restrict__ A_im) {
  const int f = blockIdx.x;
  int j = 0, rem = f;
  while (rem >= MM - j) { rem -= MM - j; ++j; }
  const int k = j + rem;
  const float scale = (j == k) ? 0.70710678118654752440f : 1.0f;

  for (int i = threadIdx.x; i < NN; i += blockDim.x) {
    const int x = i >> 5;
    const int y = i & (DD - 1);
    const float ujx_r = U_re[j * MM + x],      ujx_i = U_im[j * MM + x];
    const float uky_r = U_re[k * MM + DD + y], uky_i = U_im[k * MM + DD + y];
    const float ukx_r = U_re[k * MM + x],      ukx_i = U_im[k * MM + x];
    const float ujy_r = U_re[j * MM + DD + y], ujy_i = U_im[j * MM + DD + y];
    const float re = (ujx_r * uky_r - ujx_i * uky_i) +
                     (ukx_r * ujy_r - ukx_i * ujy_i);
    const float im = (ujx_r * uky_i + ujx_i * uky_r) +
                     (ukx_r * ujy_i + ukx_i * ujy_r);
    A_re[(size_t)f * NN + i] = re * scale;
    A_im[(size_t)f * NN + i] = im * scale;
  }
}

// ---------------------------------------------------------------------------
// Kernel 2 (fused): one workgroup per (batch b, f-tile of 16 rows).
//   Phase A: tmp[16 x 1024] = A[f-tile,:] * triu-half(rho[b])
//            (K-loop split: branch-free strictly-upper part + masked diag tail)
//   Phase B: inter[16 x 2080] = tmp * conj(A)^T, 2 g-tiles per wave,
//            Hermitian-combined into out with f32 global atomics.
// ---------------------------------------------------------------------------
extern "C" __global__ __launch_bounds__(256)
void qdense_main(const float* __restrict__ rho_re,
                 const float* __restrict__ rho_im,
                 const float* __restrict__ A_re,
                 const float* __restrict__ A_im,
                 float* __restrict__ out) {
  extern __shared__ float smem[];
  float* tmp_re = smem;             // [16][TS]
  float* tmp_im = smem + 16 * TS;   // [16][TS]

  const int ft   = blockIdx.x;      // f tile index (0..129)
  const int b    = blockIdx.y;      // batch
  const int f0   = ft * 16;
  const int lane = threadIdx.x & 31;
  const int wave = threadIdx.x >> 5;
  const int ln   = lane & 15;       // N-index within fragment
  const int hi   = lane >> 4;       // half-wave selector (K pairing)

  const size_t rbase = (size_t)b * NN * NN;

  // ---------------- Phase A: tmp strip into LDS ----------------
  const float* arow_re = A_re + (size_t)(f0 + ln) * NN;
  const float* arow_im = A_im + (size_t)(f0 + ln) * NN;

  for (int nt = wave; nt < NN / 16; nt += 8) {
    const int j0 = nt * 16;
    const int j  = j0 + ln;         // output column (B-operand N index)
    v8f Prr = {}, Qii = {}, Sri = {}, Tir = {};

    // Column pointers for rho(:, j)
    const float* rcol_re = rho_re + rbase + j;
    const float* rcol_im = rho_im + rbase + j;

    // --- main K loop: i < j0 <= j  -> strictly upper, no mask, no branches ---
#pragma unroll 4
    for (int i0 = 0; i0 < j0; i0 += 4) {
      const int ia = i0 + 2 * hi;   // this lane's K rows: ia, ia+1
      const v2f a_re = ld2(arow_re + ia);
      const v2f a_im = ld2(arow_im + ia);
      const v2f b_re = { rcol_re[(size_t)ia * NN], rcol_re[(size_t)(ia + 1) * NN] };
      const v2f b_im = { rcol_im[(size_t)ia * NN], rcol_im[(size_t)(ia + 1) * NN] };
      Prr = wmma4(a_re, b_re, Prr);
      Qii = wmma4(a_im, b_im, Qii);
      Sri = wmma4(a_re, b_im, Sri);
      Tir = wmma4(a_im, b_re, Tir);
    }

    // --- diagonal-block tail: i in [j0, j0+16), masked, diag scaled 0.5 ---
#pragma unroll
    for (int i0 = j0; i0 < j0 + 16; i0 += 4) {
      const int ia = i0 + 2 * hi;
      const v2f a_re = ld2(arow_re + ia);
      const v2f a_im = ld2(arow_im + ia);
      float br0 = 0.f, bi0 = 0.f, br1 = 0.f, bi1 = 0.f;
      if (ia <= j) {
        const float msk = (ia == j) ? 0.5f : 1.0f;
        br0 = rcol_re[(size_t)ia * NN] * msk;
        bi0 = rcol_im[(size_t)ia * NN] * msk;
      }
      if (ia + 1 <= j) {
        const float msk = (ia + 1 == j) ? 0.5f : 1.0f;
        br1 = rcol_re[(size_t)(ia + 1) * NN] * msk;
        bi1 = rcol_im[(size_t)(ia + 1) * NN] * msk;
      }
      const v2f b_re = { br0, br1 };
      const v2f b_im = { bi0, bi1 };
      Prr = wmma4(a_re, b_re, Prr);
      Qii = wmma4(a_im, b_im, Qii);
      Sri = wmma4(a_re, b_im, Sri);
      Tir = wmma4(a_im, b_re, Tir);
    }

#pragma unroll
    for (int v = 0; v < 8; ++v) {
      const int m = v + 8 * hi;     // C layout: VGPR v -> M = v (lo half) / v+8 (hi half)
      tmp_re[m * TS + j] = Prr[v] - Qii[v];
      tmp_im[m * TS + j] = Sri[v] + Tir[v];
    }
  }
  __syncthreads();

  // ---------------- Phase B: inter strip, 2 g-tiles per wave ----------------
  const float* trow_re = tmp_re + ln * TS;
  const float* trow_im = tmp_im + ln * TS;

  for (int gp = wave; gp < FT / 2; gp += 8) {
    const int g0 = gp * 32;         // two adjacent 16-wide g tiles
    const float* grow_re0 = A_re + (size_t)(g0 + ln) * NN;
    const float* grow_im0 = A_im + (size_t)(g0 + ln) * NN;
    const float* grow_re1 = A_re + (size_t)(g0 + 16 + ln) * NN;
    const float* grow_im1 = A_im + (size_t)(g0 + 16 + ln) * NN;

    v8f Prr0 = {}, Qii0 = {}, Sir0 = {}, Tri0 = {};
    v8f Prr1 = {}, Qii1 = {}, Sir1 = {}, Tri1 = {};

#pragma unroll 4
    for (int j0k = 0; j0k < NN; j0k += 4) {
      const int jk = j0k + 2 * hi;
      if ((j0k & 127) == 0) {
        __builtin_prefetch(grow_re0 + j0k + 256, 0, 0);  // global_prefetch_b8
        __builtin_prefetch(grow_im0 + j0k + 256, 0, 0);
        __builtin_prefetch(grow_re1 + j0k + 256, 0, 0);
        __builtin_prefetch(grow_im1 + j0k + 256, 0, 0);
      }
      // Shared A-operand (tmp) from LDS: reused for both g tiles
      const v2f t_re = ld2(trow_re + jk);
      const v2f t_im = ld2(trow_im + jk);
      // B operands: A[g][jk..jk+1] for the two tiles (contiguous b64, L2-hot)
      const v2f u_re0 = ld2(grow_re0 + jk);
      const v2f u_im0 = ld2(grow_im0 + jk);
      const v2f u_re1 = ld2(grow_re1 + jk);
      const v2f u_im1 = ld2(grow_im1 + jk);
      // inter = tmp * conj(A): re = Tr*Ar + Ti*Ai ; im = Ti*Ar - Tr*Ai
      Prr0 = wmma4(t_re, u_re0, Prr0);
      Qii0 = wmma4(t_im, u_im0, Qii0);
      Sir0 = wmma4(t_im, u_re0, Sir0);
      Tri0 = wmma4(t_re, u_im0, Tri0);
      Prr1 = wmma4(t_re, u_re1, Prr1);
      Qii1 = wmma4(t_im, u_im1, Qii1);
      Sir1 = wmma4(t_im, u_re1, Sir1);
      Tri1 = wmma4(t_re, u_im1, Tri1);
    }

#pragma unroll
    for (int v = 0; v < 8; ++v) {
      const int f  = f0 + v + 8 * hi;
      const int ga = g0 + ln;
      const int gb = g0 + 16 + ln;
      const float rea = Prr0[v] + Qii0[v];
      const float ima = Sir0[v] - Tri0[v];
      const float reb = Prr1[v] + Qii1[v];
      const float imb = Sir1[v] - Tri1[v];
      // new_rho = inter + conj(inter^T): scatter both images with f32 atomics
      const size_t o1a = (((size_t)b * FF + f) * FF + ga) * 2;
      const size_t o2a = (((size_t)b * FF + ga) * FF + f) * 2;
      const size_t o1b = (((size_t)b * FF + f) * FF + gb) * 2;
      const size_t o2b = (((size_t)b * FF + gb) * FF + f) * 2;
      atomicAdd(out + o1a,     rea);
      atomicAdd(out + o1a + 1, ima);
      atomicAdd(out + o2a,     rea);
      atomicAdd(out + o2a + 1, -ima);
      atomicAdd(out + o1b,     reb);
      atomicAdd(out + o1b + 1, imb);
      atomicAdd(out + o2b,     reb);
      atomicAdd(out + o2b + 1, -imb);
    }
  }
}

// ---------------------------------------------------------------------------
extern "C" void kernel_launch(void* const* d_in, const int* in_sizes, int n_in,
                              void* d_out, int out_size, void* d_ws,
                              size_t ws_size, hipStream_t stream) {
  (void)in_sizes; (void)n_in; (void)ws_size;
  const float* rho_re = (const float*)d_in[0];  // [16,1024,1024]
  const float* rho_im = (const float*)d_in[1];
  const float* U_re   = (const float*)d_in[2];  // [64,64]
  const float* U_im   = (const float*)d_in[3];

  float* out  = (float*)d_out;                  // [16,2080,2080,2]
  float* A_re = (float*)d_ws;                   // [2080,1024]
  float* A_im = A_re + (size_t)FF * NN;         // [2080,1024] (ws: ~17 MB)

  // Output is accumulated with atomics -> must start from zero every call.
  hipMemsetAsync(d_out, 0, (size_t)out_size * sizeof(float), stream);

  slos_amp_kernel<<<dim3(FF), 256, 0, stream>>>(U_re, U_im, A_re, A_im);

  const size_t lds_bytes = (size_t)(2 * 16 * TS) * sizeof(float);  // ~129 KB of 320 KB
  qdense_main<<<dim3(FT, BB), 256, lds_bytes, stream>>>(rho_re, rho_im,
                                                        A_re, A_im, out);
}